// NonParametricProjectedActor_42399917146658
// MI455X (gfx1250) — compile-verified
//
#include <hip/hip_runtime.h>
#include <hip/hip_bf16.h>
#include <math.h>

// Problem constants (from reference)
#define B_      2048
#define OBS_D   64
#define KEY_D   128
#define CAP_    131072
#define VAL_D   6
#define NW      16            // waves per block in fused kernel (512 threads)
#define TILES_CAP (CAP_ / 16) // 8192 capacity tiles of 16 slots

typedef __attribute__((ext_vector_type(2))) float v2f;
typedef __attribute__((ext_vector_type(4))) float v4f;
typedef __attribute__((ext_vector_type(8))) float v8f;

#define LOG2E 1.44269504088896340736f

// ---------------------------------------------------------------------------
// Kernel 1: h = obs @ W^T + b  (2048 x 128), and h_sq[row] = ||h_row||^2
// One block of 128 threads per batch row; thread j computes h[row][j].
// ---------------------------------------------------------------------------
__global__ __launch_bounds__(128)
void predict_kernel(const float* __restrict__ obs, const float* __restrict__ W,
                    const float* __restrict__ b, float* __restrict__ h,
                    float* __restrict__ hsq) {
    __shared__ float sobs[OBS_D];
    __shared__ float sred[128];
    const int row = blockIdx.x;
    const int j = threadIdx.x;
    if (j < OBS_D) sobs[j] = obs[row * OBS_D + j];
    __syncthreads();
    const float* wr = W + j * OBS_D;     // W is (KEY_D, OBS_D) row-major
    float s = b[j];
#pragma unroll 8
    for (int i = 0; i < OBS_D; ++i) s = fmaf(sobs[i], wr[i], s);
    h[row * KEY_D + j] = s;
    sred[j] = s * s;
    __syncthreads();
    for (int off = 64; off > 0; off >>= 1) {
        if (j < off) sred[j] += sred[j + off];
        __syncthreads();
    }
    if (j == 0) hsq[row] = sred[0];
}

// ---------------------------------------------------------------------------
// Kernel 2: k_sq[slot] = ||keys[slot]||^2   (also warms keys into L2)
// ---------------------------------------------------------------------------
__global__ __launch_bounds__(256)
void ksq_kernel(const float* __restrict__ keys, float* __restrict__ ksq) {
    const int s = blockIdx.x * blockDim.x + threadIdx.x;
    const float* kr = keys + (size_t)s * KEY_D;  // rows are 512B -> 16B aligned
    float acc = 0.f;
#pragma unroll
    for (int i = 0; i < KEY_D; i += 4) {
        v4f k = *(const v4f*)(kr + i);
        acc = fmaf(k.x, k.x, acc);
        acc = fmaf(k.y, k.y, acc);
        acc = fmaf(k.z, k.z, acc);
        acc = fmaf(k.w, k.w, acc);
    }
    ksq[s] = acc;
}

// ---------------------------------------------------------------------------
// Kernel 3: fused  2*h@keys^T - h_sq - k_sq  -> online softmax -> @values
//           -> tanh  (and broadcast std)
//
// WMMA orientation: D(16x16) = keys_tile(16 slots x K) x h^T(K x 16 rows)
//   A (16x4 f32, 2 VGPR): lane l holds slot M = l%16, K = 4*kk + 2*(l/16)+{0,1}
//   B (4x16 f32, 2 VGPR): lane l holds row  N = l%16, K = 4*kk + 2*(l/16)+{0,1}
//   C/D (16x16 f32, 8 VGPR): lane l holds col N = l%16, rows M = v + 8*(l/16)
// => batch row n's 16 slot-scores live entirely in lanes {n, n+16}; the
//    softmax-over-slots reduction is one shfl_xor(16) per tile.
// ---------------------------------------------------------------------------
__global__ __launch_bounds__(NW * 32)
void fused_softmax_kernel(const float* __restrict__ h,
                          const float* __restrict__ hsq,
                          const float* __restrict__ ksq,
                          const float* __restrict__ keys,
                          const float* __restrict__ values,
                          const float* __restrict__ log_temp,
                          const float* __restrict__ stdp,
                          float* __restrict__ out) {
    __shared__ float red[NW][16][8];   // per-wave partial (m, l, acc[6])

    const int tid  = threadIdx.x;
    const int wave = tid >> 5;
    const int lane = tid & 31;
    const int n    = lane & 15;   // batch row within tile (N column)
    const int hi   = lane >> 4;   // half-wave select
    const int row_base = blockIdx.x * 16;

    const float invT = __builtin_amdgcn_exp2f(-log_temp[0] * LOG2E); // 1/exp(lt)

    // Preload resident B operand: h^T for this 16-row tile (32 x v2f = 64 VGPR)
    v2f hb[32];
    const float* hrow = h + (size_t)(row_base + n) * KEY_D + 2 * hi;
#pragma unroll
    for (int kk = 0; kk < 32; ++kk) hb[kk] = *(const v2f*)(hrow + 4 * kk);
    const float hs = hsq[row_base + n];

    // Online softmax state (per lane; lanes n and n+16 share the same m)
    float m = -3.0e38f, lsum = 0.f;
    float acc[VAL_D] = {0.f, 0.f, 0.f, 0.f, 0.f, 0.f};

    for (int t = wave; t < TILES_CAP; t += NW) {
        const int slot_base = t * 16;
        const float* krow = keys + (size_t)(slot_base + n) * KEY_D + 2 * hi;

        // Prefetch next tile's keys row for this wave (L2 -> L0)
        __builtin_prefetch(krow + (size_t)NW * 16 * KEY_D, 0, 0);

        // Stream A operand (keys tile) and run 32 f32 WMMAs (K = 128)
        v2f a[32];
#pragma unroll
        for (int kk = 0; kk < 32; ++kk) a[kk] = *(const v2f*)(krow + 4 * kk);
        const float* kq = ksq + slot_base + 8 * hi;     // 32B aligned
        v4f kq0 = *(const v4f*)(kq);
        v4f kq1 = *(const v4f*)(kq + 4);

        v8f c = {};
#pragma unroll
        for (int kk = 0; kk < 32; ++kk)
            c = __builtin_amdgcn_wmma_f32_16x16x4_f32(false, a[kk], false, hb[kk],
                                                      (short)0, c, false, false);

        // logits = (2*dot - |h|^2 - |k|^2) / T ; c[v] is slot v + 8*hi
        float lg[8];
#pragma unroll
        for (int v = 0; v < 8; ++v) {
            const float kv = (v < 4) ? kq0[v] : kq1[v - 4];
            lg[v] = (2.0f * c[v] - hs - kv) * invT;
        }

        // tile max across this lane's 8 slots + partner lane's 8 slots
        float tmax = fmaxf(fmaxf(fmaxf(lg[0], lg[1]), fmaxf(lg[2], lg[3])),
                           fmaxf(fmaxf(lg[4], lg[5]), fmaxf(lg[6], lg[7])));
        tmax = fmaxf(tmax, __shfl_xor(tmax, 16, 32));
        const float mn = fmaxf(m, tmax);
        const float scale = __builtin_amdgcn_exp2f((m - mn) * LOG2E);
        lsum *= scale;
#pragma unroll
        for (int d = 0; d < VAL_D; ++d) acc[d] *= scale;

#pragma unroll
        for (int v = 0; v < 8; ++v) {
            const float w = __builtin_amdgcn_exp2f((lg[v] - mn) * LOG2E);
            lsum += w;
            const float* vp = values + (size_t)(slot_base + 8 * hi + v) * VAL_D;
            v2f v0 = *(const v2f*)(vp);       // rows are 24B -> 8B aligned
            v2f v1 = *(const v2f*)(vp + 2);
            v2f v2 = *(const v2f*)(vp + 4);
            acc[0] = fmaf(w, v0.x, acc[0]);
            acc[1] = fmaf(w, v0.y, acc[1]);
            acc[2] = fmaf(w, v1.x, acc[2]);
            acc[3] = fmaf(w, v1.y, acc[3]);
            acc[4] = fmaf(w, v2.x, acc[4]);
            acc[5] = fmaf(w, v2.y, acc[5]);
        }
        m = mn;
    }

    // Merge the lane pair (m is already identical in both halves)
    lsum += __shfl_xor(lsum, 16, 32);
#pragma unroll
    for (int d = 0; d < VAL_D; ++d) acc[d] += __shfl_xor(acc[d], 16, 32);

    if (hi == 0) {
        red[wave][n][0] = m;
        red[wave][n][1] = lsum;
#pragma unroll
        for (int d = 0; d < VAL_D; ++d) red[wave][n][2 + d] = acc[d];
    }
    __syncthreads();

    // Wave 0, lanes 0..15: merge the NW partial softmaxes, finalize, store
    if (tid < 16) {
        float M = red[0][tid][0];
        float L = red[0][tid][1];
        float A[VAL_D];
#pragma unroll
        for (int d = 0; d < VAL_D; ++d) A[d] = red[0][tid][2 + d];
        for (int w = 1; w < NW; ++w) {
            const float m2 = red[w][tid][0], l2 = red[w][tid][1];
            const float Mn = fmaxf(M, m2);
            const float s1 = __builtin_amdgcn_exp2f((M - Mn) * LOG2E);
            const float s2 = __builtin_amdgcn_exp2f((m2 - Mn) * LOG2E);
            L = L * s1 + l2 * s2;
#pragma unroll
            for (int d = 0; d < VAL_D; ++d)
                A[d] = A[d] * s1 + red[w][tid][2 + d] * s2;
            M = Mn;
        }
        const float inv = 1.0f / L;
        const float s0 = stdp[0];
        const int row = row_base + tid;
#pragma unroll
        for (int d = 0; d < VAL_D; ++d) {
            out[row * VAL_D + d] = tanhf(A[d] * inv);             // mu
            out[B_ * VAL_D + row * VAL_D + d] = s0;               // std_out
        }
    }
}

// ---------------------------------------------------------------------------
extern "C" void kernel_launch(void* const* d_in, const int* in_sizes, int n_in,
                              void* d_out, int out_size, void* d_ws, size_t ws_size,
                              hipStream_t stream) {
    const float* obs      = (const float*)d_in[0];   // (B, OBS_D)
    const float* stdp     = (const float*)d_in[1];   // (1,)
    const float* W        = (const float*)d_in[2];   // (KEY_D, OBS_D)
    const float* b        = (const float*)d_in[3];   // (KEY_D,)
    const float* keys     = (const float*)d_in[4];   // (CAP, KEY_D)
    const float* values   = (const float*)d_in[5];   // (CAP, VAL_D)
    const float* log_temp = (const float*)d_in[6];   // (,)

    float* out = (float*)d_out;                      // mu (B*6) ++ std (B*6)

    // Workspace layout (floats): h (B*KEY_D) | hsq (B) | ksq (CAP)
    float* h   = (float*)d_ws;
    float* hsq = h + (size_t)B_ * KEY_D;
    float* ksq = hsq + B_;

    predict_kernel<<<B_, 128, 0, stream>>>(obs, W, b, h, hsq);
    ksq_kernel<<<CAP_ / 256, 256, 0, stream>>>(keys, ksq);
    fused_softmax_kernel<<<B_ / 16, NW * 32, 0, stream>>>(
        h, hsq, ksq, keys, values, log_temp, stdp, out);
}